// ConvCaps_30734785970324
// MI455X (gfx1250) — compile-verified
//
#include <hip/hip_runtime.h>
#include <hip/hip_bf16.h>

// ConvCaps EM-routing capsule conv for gfx1250 (MI455X).
//
// Kernel 1: pose transform v[n,k,c] = P_in[n,k](4x4) @ W[k,c](4x4) via
//   V_WMMA_F32_16X16X4_F32. Per k' (288 of them) the GEMM is M=1568, N=128,
//   K=4. One block per (k', M-slice): B panel (2 KB) staged to LDS once
//   via GLOBAL_LOAD_ASYNC_TO_LDS_B128 + s_wait_asynccnt, each wave holds all
//   8 B fragments in registers and does 8 WMMAs per M-tile -> 8x less
//   overhead per WMMA than one-tile-per-wave. D stored as two float4 per
//   tile per lane into layout v[n][k][col][i] (col=c*4+l) -> 256 B
//   contiguous per half-wave.
// Kernel 2: EM routing, one 256-thread workgroup per n (392 blocks).
//   Each block re-reads its 576 KB v-slice ~8x; slice stays resident in the
//   192 MB L2, so HBM traffic ~= 1 write + 1 read of the 231 MB v (~20us).
//
// Workspace: v = 392*288*128*4 f32 = 231 MB at d_ws.

#define C_CAP   32
#define NKB     288            // K*K*B = 3*3*32
#define N_POS   392            // 8*7*7
#define MTILES  98             // (392*4)/16
#define XCH     544            // 32*17 channels per pixel
#define MT_PER_BLK 14          // 98 / 7 y-blocks

typedef __attribute__((ext_vector_type(2))) float v2f;
typedef __attribute__((ext_vector_type(8))) float v8f;
typedef int v4i __attribute__((ext_vector_type(4)));

#if defined(__has_builtin)
#if __has_builtin(__builtin_amdgcn_global_load_async_to_lds_b128) && \
    __has_builtin(__builtin_amdgcn_s_wait_asynccnt)
#define USE_ASYNC_LDS 1
#endif
#endif
#ifndef USE_ASYNC_LDS
#define USE_ASYNC_LDS 0
#endif

#if USE_ASYNC_LDS
typedef __attribute__((address_space(1))) v4i gv4i;   // global int4
typedef __attribute__((address_space(3))) v4i lv4i;   // LDS int4
#endif

// ---------------------------------------------------------------------------
// Kernel 1: pose transform with fp32 WMMA (16x16x4), 8 WMMAs per M-tile.
// A frag (ISA 7.12.2, 32-bit A 16x4): lanes 0-15 hold K=0,1 in v[0],v[1];
//   lanes 16-31 hold K=2,3.  B (4x16) assumed mirrored (rows striped across
//   lanes, K split across lane halves).  C/D: VGPR r -> M=r (lanes 0-15),
//   M=r+8 (lanes 16-31), N = lane&15.
// v layout: [n][k'][col][i], col = c*4+l, i = pose row -> b128 stores,
//   256 B contiguous per half-wave.
// ---------------------------------------------------------------------------
__global__ __launch_bounds__(256) void pose_wmma_kernel(
    const float* __restrict__ x,      // (8,16,16,544)
    const float* __restrict__ w,      // (288,32,4,4)
    float* __restrict__ v_out) {      // (392,288,128,4)
  __shared__ float Bs[512];           // B panel for this k' (2 KB)

  const int lane   = threadIdx.x & 31;
  const int wave   = threadIdx.x >> 5;
  const int kprime = blockIdx.x;      // 0..287
  const int mt0    = blockIdx.y * MT_PER_BLK;

  const int kh   = kprime / 96;
  const int kw   = (kprime / 32) % 3;
  const int bcap = kprime & 31;

  // ---- stage B panel (w[k',*,*,*], 512 f32) into LDS --------------------
#if USE_ASYNC_LDS
  if (threadIdx.x < 32) {             // wave 0: 4 x (32 lanes x 16B) = 2 KB
    const float* gsrc = w + (long)kprime * 512 + lane * 4;
#pragma unroll
    for (int itb = 0; itb < 4; ++itb) {
      __builtin_amdgcn_global_load_async_to_lds_b128(
          (gv4i*)(unsigned long long)(gsrc + itb * 128),
          (lv4i*)(unsigned long long)(&Bs[lane * 4 + itb * 128]),
          0, 0);
    }
    __builtin_amdgcn_s_wait_asynccnt(0);
  }
#else
  for (int t = threadIdx.x; t < 512; t += 256)
    Bs[t] = w[(long)kprime * 512 + t];
#endif
  __syncthreads();

  // ---- build all 8 B fragments (N = 0..127) from LDS --------------------
  v2f bfrag[8];
  const int jj = (lane >> 4) << 1;    // K offset: 0 or 2
#pragma unroll
  for (int nt = 0; nt < 8; ++nt) {
    const int col = nt * 16 + (lane & 15);            // c*4 + l
    const int bi  = ((col >> 2) << 4) + (jj << 2) + (col & 3);
    bfrag[nt].x = Bs[bi];
    bfrag[nt].y = Bs[bi + 4];
  }

  // ---- loop over this wave's M-tiles ------------------------------------
  for (int mt = mt0 + wave; mt < mt0 + MT_PER_BLK; mt += 8) {
    // A fragment: A[m, j] = p_in[n, k', i, j], m = n*4 + i
    const int arow = mt * 16 + (lane & 15);
    const int nidx = arow >> 2;
    const int ii   = arow & 3;
    const int bb   = nidx / 49;
    const int rr2  = nidx - bb * 49;
    const int oi   = rr2 / 7;
    const int oj   = rr2 - oi * 7;
    const long xoff = ((long)(bb * 16 + 2 * oi + kh) * 16 + (2 * oj + kw)) * XCH
                      + bcap * 16 + ii * 4 + jj;
    v2f a;
    a.x = x[xoff];
    a.y = x[xoff + 1];

    v8f acc[8];
#pragma unroll
    for (int nt = 0; nt < 8; ++nt) {
      acc[nt] = (v8f){};
      acc[nt] = __builtin_amdgcn_wmma_f32_16x16x4_f32(
          false, a, false, bfrag[nt], (short)0, acc[nt], false, false);
    }

    // D stores: VGPRs 0-3 = (n0, i=0..3), VGPRs 4-7 = (n0+1, i=0..3)
    const int n0 = mt * 4 + ((lane >> 4) << 1);
#pragma unroll
    for (int nt = 0; nt < 8; ++nt) {
      const int col = nt * 16 + (lane & 15);
      float* p0 = v_out + (((long)n0 * NKB + kprime) * 128 + col) * 4;
      float* p1 = v_out + ((((long)n0 + 1) * NKB + kprime) * 128 + col) * 4;
      *(float4*)p0 = make_float4(acc[nt][0], acc[nt][1], acc[nt][2], acc[nt][3]);
      *(float4*)p1 = make_float4(acc[nt][4], acc[nt][5], acc[nt][6], acc[nt][7]);
    }
  }
}

// ---------------------------------------------------------------------------
// Kernel 2: EM routing, one workgroup per spatial position n.
// v layout per n: vb[k*512 + col*4 + i], col = c*4+l, e = i*4+l.
// ---------------------------------------------------------------------------
__global__ __launch_bounds__(256) void em_routing_kernel(
    const float* __restrict__ x,
    const float* __restrict__ v,
    const float* __restrict__ beta_a,
    const float* __restrict__ beta_u,
    float* __restrict__ out) {
  __shared__ float r_s[NKB * C_CAP];   // 36 KB: r / rr / ln_ap (reused)
  __shared__ float a_in[NKB];
  __shared__ float mu_s[512];          // indexed c*16 + e
  __shared__ float sig_s[512];
  __shared__ float lsg_s[512];
  __shared__ float r_sum[C_CAP];
  __shared__ float inv_rs[C_CAP];
  __shared__ float a_out_s[C_CAP];

  const int n   = blockIdx.x;          // < 392
  const int tid = threadIdx.x;
  const int bb  = n / 49;
  const int rr2 = n - bb * 49;
  const int oi  = rr2 / 7;
  const int oj  = rr2 - oi * 7;

  for (int k = tid; k < NKB; k += 256) {
    const int kh = k / 96, kw = (k / 32) % 3, bcap = k & 31;
    const long xoff =
        ((long)(bb * 16 + 2 * oi + kh) * 16 + (2 * oj + kw)) * XCH + 512 + bcap;
    a_in[k] = x[xoff];
  }
  for (int q = tid; q < NKB * C_CAP; q += 256) r_s[q] = 1.0f / C_CAP;
  __syncthreads();

  const float* vb = v + (long)n * NKB * 512;

  for (int it = 0; it < 3; ++it) {
    // -- rr = r*a_in, normalized over c (each thread owns whole rows) -----
    for (int k = tid; k < NKB; k += 256) {
      const float ai = a_in[k];
      float tmp[C_CAP];
      float s = 0.f;
#pragma unroll
      for (int c2 = 0; c2 < C_CAP; ++c2) { tmp[c2] = r_s[k * C_CAP + c2] * ai; s += tmp[c2]; }
      const float inv = 1.f / (s + 1e-8f);
#pragma unroll
      for (int c2 = 0; c2 < C_CAP; ++c2) r_s[k * C_CAP + c2] = tmp[c2] * inv;
    }
    if (tid < C_CAP) r_sum[tid] = 0.f;
    __syncthreads();

    // -- r_sum[c] = sum_k rr[k,c] (8 partials per c, LDS float atomics) ---
    {
      const int c2 = tid & 31, g = tid >> 5;
      float p = 0.f;
      for (int k = g; k < NKB; k += 8) p += r_s[k * C_CAP + c2];
      atomicAdd(&r_sum[c2], p);
    }
    __syncthreads();
    if (tid < C_CAP) inv_rs[tid] = 1.f / (r_sum[tid] + 1e-8f);
    __syncthreads();

    // -- mu[c,e] = sum_k coeff*v (coalesced: consecutive tid -> consec v) -
    for (int p = tid; p < 512; p += 256) {
      const int i  = p & 3;
      const int l  = (p >> 2) & 3;
      const int c2 = p >> 4;
      const int me = c2 * 16 + i * 4 + l;
      const float irs = inv_rs[c2];
      float acc = 0.f;
      for (int k = 0; k < NKB; ++k)
        acc += r_s[k * C_CAP + c2] * irs * vb[k * 512 + p];
      mu_s[me] = acc;
    }
    __syncthreads();

    // -- sigma_sq, log_sigma ----------------------------------------------
    for (int p = tid; p < 512; p += 256) {
      const int i  = p & 3;
      const int l  = (p >> 2) & 3;
      const int c2 = p >> 4;
      const int me = c2 * 16 + i * 4 + l;
      const float irs = inv_rs[c2];
      const float m = mu_s[me];
      float acc = 0.f;
      for (int k = 0; k < NKB; ++k) {
        const float d = vb[k * 512 + p] - m;
        acc += r_s[k * C_CAP + c2] * irs * d * d;
      }
      const float ss = acc + 1e-8f;
      sig_s[me] = ss;
      lsg_s[me] = 0.5f * __logf(ss);
    }
    __syncthreads();

    // -- a_out[c] ---------------------------------------------------------
    if (tid < C_CAP) {
      float cost = 0.f;
#pragma unroll
      for (int e = 0; e < 16; ++e) cost += beta_u[tid] + lsg_s[tid * 16 + e];
      cost *= r_sum[tid];
      a_out_s[tid] = 1.f / (1.f + __expf(-0.001f * (beta_a[tid] - cost)));
    }
    __syncthreads();

    if (it < 2) {
      // -- ln_ap[k,c] = sum_e ln_p + log(a_out); per (k,c) the 16 v values
      //    are a contiguous 64 B chunk. Overwrites r_s. --------------------
      for (int q = tid; q < NKB * C_CAP; q += 256) {
        const int k = q >> 5, c2 = q & 31;
        const float* vk = vb + k * 512 + c2 * 16;   // [l*4 + i]
        float acc = 0.f;
#pragma unroll
        for (int l = 0; l < 4; ++l)
#pragma unroll
          for (int i = 0; i < 4; ++i) {
            const int e = i * 4 + l;
            const float d = vk[l * 4 + i] - mu_s[c2 * 16 + e];
            acc += -d * d / (2.f * sig_s[c2 * 16 + e]) - lsg_s[c2 * 16 + e];
          }
        acc += __logf(a_out_s[c2]) - 14.703517f;  // 16 * 0.5*ln(2*pi)
        r_s[q] = acc;
      }
      __syncthreads();
      // -- softmax over c per row k ---------------------------------------
      for (int k = tid; k < NKB; k += 256) {
        float mx = -3.4e38f;
#pragma unroll
        for (int c2 = 0; c2 < C_CAP; ++c2) mx = fmaxf(mx, r_s[k * C_CAP + c2]);
        float s = 0.f;
#pragma unroll
        for (int c2 = 0; c2 < C_CAP; ++c2) {
          const float t = __expf(r_s[k * C_CAP + c2] - mx);
          r_s[k * C_CAP + c2] = t;
          s += t;
        }
        const float inv = 1.f / s;
#pragma unroll
        for (int c2 = 0; c2 < C_CAP; ++c2) r_s[k * C_CAP + c2] *= inv;
      }
      __syncthreads();
    }
  }

  // -- write p_out (mu) and a_out: (n, 512 + 32) --------------------------
  for (int p = tid; p < 512; p += 256) out[(long)n * XCH + p] = mu_s[p];
  if (tid < C_CAP) out[(long)n * XCH + 512 + tid] = a_out_s[tid];
}

extern "C" void kernel_launch(void* const* d_in, const int* in_sizes, int n_in,
                              void* d_out, int out_size, void* d_ws, size_t ws_size,
                              hipStream_t stream) {
  (void)in_sizes; (void)n_in; (void)out_size; (void)ws_size;
  const float* x      = (const float*)d_in[0];  // (8,16,16,544)
  const float* w      = (const float*)d_in[1];  // (1,288,32,4,4)
  const float* beta_a = (const float*)d_in[2];  // (32,)
  const float* beta_u = (const float*)d_in[3];  // (32,)
  float* out = (float*)d_out;                   // (8,7,7,544)
  float* v   = (float*)d_ws;                    // 392*288*128*4 f32 = 231 MB

  pose_wmma_kernel<<<dim3(NKB, MTILES / MT_PER_BLK), 256, 0, stream>>>(x, w, v);
  em_routing_kernel<<<N_POS, 256, 0, stream>>>(x, v, beta_a, beta_u, out);
}